// FocalLoss_62904091017770
// MI455X (gfx1250) — compile-verified
//
#include <hip/hip_runtime.h>
#include <hip/hip_bf16.h>

#define TPB 256

typedef __attribute__((ext_vector_type(2))) float v2f;
typedef __attribute__((ext_vector_type(8))) float v8f;

#if defined(__has_builtin)
#if __has_builtin(__builtin_amdgcn_wmma_f32_16x16x4_f32)
#define HAVE_WMMA_F32_K4 1
#endif
#endif

// ---------------- helpers ----------------

__device__ __forceinline__ float smooth_l1(float d) {
    float ad = fabsf(d);
    return (ad < 1.0f) ? 0.5f * d * d : ad - 0.5f;
}

// Deterministic block tree reduction (safe for repeated use: entry barrier).
__device__ __forceinline__ float block_reduce_sum(float v, float* sm) {
    int tid = threadIdx.x;
    __syncthreads();
    sm[tid] = v;
    __syncthreads();
    for (int s = TPB / 2; s > 0; s >>= 1) {
        if (tid < s) sm[tid] += sm[tid + s];
        __syncthreads();
    }
    return sm[0];
}

// ---------------- kernel 0: zero best-prior keys ----------------

__global__ void init_bp_kernel(unsigned long long* bp, int n) {
    int i = blockIdx.x * blockDim.x + threadIdx.x;
    if (i < n) bp[i] = 0ull;
}

// ---------------- kernel 1: per-truth argmax over anchors ----------------
// key = (iou_bits << 32) | (0xFFFFFFFF - anchor) : max-key == max IoU with
// first-index (lowest anchor) tie-break, matching jnp.argmax. atomicMax is
// commutative -> deterministic.

__global__ void best_prior_kernel(const float* __restrict__ truths,   // [B,16,4] point form
                                  const float* __restrict__ anchors,  // [A,4] center form
                                  unsigned long long* __restrict__ bp,// [B*16]
                                  int A, int blocksPerBatch) {
    __shared__ float tr[64];
    __shared__ unsigned long long keys[16 * TPB];
    __shared__ unsigned long long red[256];
    int tid = threadIdx.x;
    int b = blockIdx.x / blocksPerBatch;
    int a = (blockIdx.x % blocksPerBatch) * TPB + tid;
    if (tid < 64) tr[tid] = truths[b * 64 + tid];
    __syncthreads();

    float acx = 0.f, acy = 0.f, aw = 1.f, ah = 1.f;
    bool valid = (a < A);
    if (valid) {
        acx = anchors[a * 4 + 0]; acy = anchors[a * 4 + 1];
        aw  = anchors[a * 4 + 2]; ah  = anchors[a * 4 + 3];
    }
    float px0 = acx - 0.5f * aw, py0 = acy - 0.5f * ah;
    float px1 = acx + 0.5f * aw, py1 = acy + 0.5f * ah;
    float areap = aw * ah;

    #pragma unroll
    for (int n = 0; n < 16; ++n) {
        float t0 = tr[n * 4 + 0], t1 = tr[n * 4 + 1];
        float t2 = tr[n * 4 + 2], t3 = tr[n * 4 + 3];
        float iw = fminf(t2, px1) - fmaxf(t0, px0);
        float ih = fminf(t3, py1) - fmaxf(t1, py0);
        iw = fmaxf(iw, 0.f); ih = fmaxf(ih, 0.f);
        float inter = iw * ih;
        float areat = (t2 - t0) * (t3 - t1);
        float iou = inter / (areat + areap - inter);
        unsigned long long key = 0ull;
        if (valid)
            key = ((unsigned long long)__float_as_uint(iou) << 32)
                | (unsigned long long)(0xFFFFFFFFu - (unsigned)a);
        keys[n * TPB + tid] = key;
    }
    __syncthreads();
    {   // 16 parallel reductions: 16 threads per truth
        int n = tid >> 4, j = tid & 15;
        unsigned long long m = keys[n * TPB + j * 16];
        #pragma unroll
        for (int k = 1; k < 16; ++k) {
            unsigned long long v = keys[n * TPB + j * 16 + k];
            m = (v > m) ? v : m;
        }
        red[n * 16 + j] = m;
    }
    __syncthreads();
    if (tid < 16) {
        unsigned long long m = red[tid * 16];
        #pragma unroll
        for (int k = 1; k < 16; ++k) {
            unsigned long long v = red[tid * 16 + k];
            m = (v > m) ? v : m;
        }
        atomicMax(&bp[b * 16 + tid], m);
    }
}

// ---------------- kernel 2: per-anchor match + smooth-L1 ----------------

__global__ void match_loc_kernel(const float* __restrict__ loc_preds, // [B,A,4]
                                 const float* __restrict__ truths,    // [B,16,4]
                                 const float* __restrict__ anchors,   // [A,4]
                                 const int* __restrict__ labels,      // [B,16]
                                 const unsigned long long* __restrict__ bp,
                                 int* __restrict__ conf_out,          // [B*A]
                                 float* __restrict__ loc_part,        // [gridDim]
                                 float* __restrict__ cnt_part,        // [gridDim]
                                 int A, int blocksPerBatch) {
    __shared__ float tr[64];
    __shared__ int lab[16];
    __shared__ unsigned bpa[16];
    __shared__ float sm[TPB];
    int tid = threadIdx.x;
    int b = blockIdx.x / blocksPerBatch;
    int a = (blockIdx.x % blocksPerBatch) * TPB + tid;
    if (tid < 64) tr[tid] = truths[b * 64 + tid];
    if (tid < 16) {
        lab[tid] = labels[b * 16 + tid];
        bpa[tid] = 0xFFFFFFFFu - (unsigned)(bp[b * 16 + tid] & 0xFFFFFFFFull);
    }
    __syncthreads();

    float ll = 0.f, cnt = 0.f;
    if (a < A) {
        float acx = anchors[a * 4 + 0], acy = anchors[a * 4 + 1];
        float aw  = anchors[a * 4 + 2], ah  = anchors[a * 4 + 3];
        float px0 = acx - 0.5f * aw, py0 = acy - 0.5f * ah;
        float px1 = acx + 0.5f * aw, py1 = acy + 0.5f * ah;
        float areap = aw * ah;

        float best = -1.f; int bestn = 0;
        #pragma unroll
        for (int n = 0; n < 16; ++n) {
            float t0 = tr[n * 4 + 0], t1 = tr[n * 4 + 1];
            float t2 = tr[n * 4 + 2], t3 = tr[n * 4 + 3];
            float iw = fminf(t2, px1) - fmaxf(t0, px0);
            float ih = fminf(t3, py1) - fmaxf(t1, py0);
            iw = fmaxf(iw, 0.f); ih = fmaxf(ih, 0.f);
            float inter = iw * ih;
            float areat = (t2 - t0) * (t3 - t1);
            float iou = inter / (areat + areap - inter);
            if (iou > best) { best = iou; bestn = n; }   // first-max tie-break
        }
        // override: every truth claims its best prior (ascending -> last wins,
        // matching scatter semantics)
        #pragma unroll
        for (int n = 0; n < 16; ++n) {
            if ((unsigned)a == bpa[n]) { best = 2.0f; bestn = n; }
        }
        int cf = (best < 0.5f) ? 0 : lab[bestn];
        int row = b * A + a;
        conf_out[row] = cf;
        if (cf > 0) {
            cnt = 1.f;
            float t0 = tr[bestn * 4 + 0], t1 = tr[bestn * 4 + 1];
            float t2 = tr[bestn * 4 + 2], t3 = tr[bestn * 4 + 3];
            float gcx = ((t0 + t2) * 0.5f - acx) / (0.1f * aw);
            float gcy = ((t1 + t3) * 0.5f - acy) / (0.1f * ah);
            float gw  = logf((t2 - t0) / aw) / 0.2f;
            float gh  = logf((t3 - t1) / ah) / 0.2f;
            float4 lp = *(const float4*)(loc_preds + (size_t)row * 4);
            ll = smooth_l1(lp.x - gcx) + smooth_l1(lp.y - gcy)
               + smooth_l1(lp.z - gw)  + smooth_l1(lp.w - gh);
        }
    }
    float s1 = block_reduce_sum(ll, sm);
    float s2 = block_reduce_sum(cnt, sm);
    if (tid == 0) { loc_part[blockIdx.x] = s1; cnt_part[blockIdx.x] = s2; }
}

// ---------------- kernel 3: focal loss sweep (bandwidth-bound) ----------------

__global__ void focal_kernel(const float* __restrict__ cp,   // [B*A*21] flat
                             const int* __restrict__ conf,   // [B*A]
                             float* __restrict__ cls_part,   // [gridDim]
                             unsigned total4) {
    const float4* cp4 = (const float4*)cp;
    __shared__ float sm[TPB];
    unsigned stride = gridDim.x * blockDim.x;
    float acc = 0.f;
    for (unsigned i = blockIdx.x * blockDim.x + threadIdx.x; i < total4; i += stride) {
        __builtin_prefetch(cp4 + i + stride, 0, 1);  // global_prefetch_b8
        float4 v = cp4[i];
        float e0 = v.x, e1 = v.y, e2 = v.z, e3 = v.w;
        float el[4] = {e0, e1, e2, e3};
        unsigned base = i * 4u;
        #pragma unroll
        for (int j = 0; j < 4; ++j) {
            unsigned flat = base + (unsigned)j;
            unsigned c   = flat % 21u;
            unsigned row = flat / 21u;
            if (c != 0u) {                      // drop background column
                float x = el[j];
                bool t = ((unsigned)conf[row] == c);
                float p  = 1.f / (1.f + expf(-x));
                float pt = t ? p : (1.f - p);
                float om = 1.f - pt;
                float w  = om * om * (t ? 0.25f : 0.75f);
                float bce = fmaxf(x, 0.f) - (t ? x : 0.f) + log1pf(expf(-fabsf(x)));
                acc += w * bce;
            }
        }
    }
    float s = block_reduce_sum(acc, sm);
    if (threadIdx.x == 0) cls_part[blockIdx.x] = s;
}

// ---------------- kernel 4: one-wave finalize with WMMA reduction ----------------
// Sum of n floats by one wave32: each lane accumulates a strided pair, the 64
// accumulators are fed as the B operand (4x16) of v_wmma_f32_16x16x4_f32 with
// A = all-ones, so D[m][n] = colsum(n); lanes 0-15 of D VGPR0 hold the 16
// column sums; shfl-xor tree finishes. Exact f32, deterministic.

__device__ float wave_sum32(const float* __restrict__ p, int n) {
    int lane = threadIdx.x;   // block has exactly 32 threads
    v2f acc; acc.x = 0.f; acc.y = 0.f;
    for (int i = 2 * lane; i < n; i += 64) {
        acc.x += p[i];
        if (i + 1 < n) acc.y += p[i + 1];
    }
#ifdef HAVE_WMMA_F32_K4
    v2f ones; ones.x = 1.f; ones.y = 1.f;
    v8f cz = {};
    v8f d = __builtin_amdgcn_wmma_f32_16x16x4_f32(
        false, ones, false, acc, (short)0, cz, false, false);
    float v = (lane < 16) ? d[0] : 0.f;
#else
    float v = acc.x + acc.y;
#endif
    v += __shfl_xor(v, 16, 32);
    v += __shfl_xor(v, 8, 32);
    v += __shfl_xor(v, 4, 32);
    v += __shfl_xor(v, 2, 32);
    v += __shfl_xor(v, 1, 32);
    return v;
}

__global__ void finalize_kernel(const float* __restrict__ cls_part, int nc,
                                const float* __restrict__ loc_part,
                                const float* __restrict__ cnt_part, int nb,
                                float* __restrict__ out) {
    float cls = wave_sum32(cls_part, nc);
    float loc = wave_sum32(loc_part, nb);
    float cnt = wave_sum32(cnt_part, nb);
    if (threadIdx.x == 0) {
        float inv = 1.f / cnt;
        float c = cls * inv, l = loc * inv;
        out[0] = c;
        out[1] = l;
        out[2] = c + l;
    }
}

// ---------------- launch ----------------

extern "C" void kernel_launch(void* const* d_in, const int* in_sizes, int n_in,
                              void* d_out, int out_size, void* d_ws, size_t ws_size,
                              hipStream_t stream) {
    const float* class_preds = (const float*)d_in[0];  // [B,A,21]
    const float* loc_preds   = (const float*)d_in[1];  // [B,A,4]
    const float* loc_targets = (const float*)d_in[2];  // [B,16,4]
    const float* anchors     = (const float*)d_in[3];  // [A,4]
    const int*   class_tg    = (const int*)d_in[4];    // [B,16]
    float* out = (float*)d_out;

    int A = in_sizes[3] / 4;
    int B = in_sizes[4] / 16;
    int blocksPerBatch = (A + TPB - 1) / TPB;
    int nBlocksBA = B * blocksPerBatch;          // 3072 for B=32, A=24576
    const int NBC = 2048;                        // focal partials (mult of 64)

    // workspace carve-up (all slots rewritten every call)
    char* ws = (char*)d_ws;
    size_t off = 0;
    unsigned long long* bp = (unsigned long long*)(ws + off);
    off += (size_t)B * 16 * sizeof(unsigned long long); off = (off + 255) & ~(size_t)255;
    int* conf = (int*)(ws + off);
    off += (size_t)B * A * sizeof(int);                 off = (off + 255) & ~(size_t)255;
    float* loc_part = (float*)(ws + off);
    off += (size_t)nBlocksBA * sizeof(float);           off = (off + 255) & ~(size_t)255;
    float* cnt_part = (float*)(ws + off);
    off += (size_t)nBlocksBA * sizeof(float);           off = (off + 255) & ~(size_t)255;
    float* cls_part = (float*)(ws + off);

    init_bp_kernel<<<(B * 16 + 255) / 256, 256, 0, stream>>>(bp, B * 16);
    best_prior_kernel<<<nBlocksBA, TPB, 0, stream>>>(loc_targets, anchors, bp,
                                                     A, blocksPerBatch);
    match_loc_kernel<<<nBlocksBA, TPB, 0, stream>>>(loc_preds, loc_targets, anchors,
                                                    class_tg, bp, conf,
                                                    loc_part, cnt_part,
                                                    A, blocksPerBatch);
    unsigned total4 = (unsigned)B * (unsigned)A * 21u / 4u;
    focal_kernel<<<NBC, TPB, 0, stream>>>(class_preds, conf, cls_part, total4);
    finalize_kernel<<<1, 32, 0, stream>>>(cls_part, NBC, loc_part, cnt_part,
                                          nBlocksBA, out);
}